// BPFingerprint_7919919693922
// MI455X (gfx1250) — compile-verified
//
#include <hip/hip_runtime.h>

#define THREADS 256
#define TILE_ATOMS 64  // atoms staged per block in the async path (12 KB LDS)

// --------------------------------------------------------------------------
// Feature probe: gfx1250 async global->LDS builtins (ASYNCcnt machinery).
// Probe-confirmed this round: builtin exists, takes 4 args, param0 is
// 'int __vector(4) __device__ *' (AS1 int4*). Param1 is the LDS-side int4*.
// --------------------------------------------------------------------------
#if defined(__has_builtin)
#if __has_builtin(__builtin_amdgcn_global_load_async_to_lds_b128) && \
    __has_builtin(__builtin_amdgcn_s_wait_asynccnt)
#define HAVE_ASYNC_LDS 1
#endif
#endif
#ifndef HAVE_ASYNC_LDS
#define HAVE_ASYNC_LDS 0
#endif

#if HAVE_ASYNC_LDS
typedef int v4i __attribute__((ext_vector_type(4)));
typedef __attribute__((address_space(1))) v4i* gv4i_p;  // global int4*
typedef __attribute__((address_space(3))) v4i* lv4i_p;  // LDS int4*
#endif

// --------------------------------------------------------------------------
// wave32 helpers
// --------------------------------------------------------------------------
__device__ __forceinline__ unsigned lane_mask_lt() {
    return (1u << (threadIdx.x & 31)) - 1u;  // lane 31 -> 0x7fffffff, no UB
}

// ---------------------------------------------------------------------------
// Pass 1: per-block count of atoms with elems == 1 (ballot + popc, 1 barrier).
// ---------------------------------------------------------------------------
__global__ void bp_count(const int* __restrict__ elems, int n,
                         int* __restrict__ partial) {
    __shared__ int wsum[THREADS / 32];
    const int tid = threadIdx.x;
    const int i = blockIdx.x * THREADS + tid;
    const int flag = (i < n && elems[i] == 1) ? 1 : 0;
    const unsigned m = (unsigned)__ballot(flag != 0);
    if ((tid & 31) == 0) wsum[tid >> 5] = __popc(m);
    __syncthreads();
    if (tid == 0) {
        int s = 0;
        for (int w = 0; w < THREADS / 32; ++w) s += wsum[w];
        partial[blockIdx.x] = s;
    }
}

// ---------------------------------------------------------------------------
// Pass 2: single-block exclusive scan of per-block counts (runs once, tiny).
// partial[0..g-1] -> exclusive offsets, partial[g] = total H count.
// ---------------------------------------------------------------------------
__global__ void bp_scan_partials(int* __restrict__ partial, int g) {
    __shared__ int sdata[THREADS];
    __shared__ int s_run;
    const int tid = threadIdx.x;
    if (tid == 0) s_run = 0;
    __syncthreads();
    for (int base = 0; base < g; base += THREADS) {
        const int i = base + tid;
        const int v = (i < g) ? partial[i] : 0;
        sdata[tid] = v;
        __syncthreads();
        for (int off = 1; off < THREADS; off <<= 1) {
            const int t = (tid >= off) ? sdata[tid - off] : 0;
            __syncthreads();
            sdata[tid] += t;
            __syncthreads();
        }
        if (i < g) partial[i] = s_run + sdata[tid] - v;  // exclusive prefix
        __syncthreads();
        if (tid == THREADS - 1) s_run += sdata[THREADS - 1];
        __syncthreads();
    }
    if (tid == 0) partial[g] = s_run;  // nH total
}

// ---------------------------------------------------------------------------
// Pass 3: per-atom output row via stable partition (wave32 ballot scan):
//   H atom -> (# H before it);  O atom -> nH + i - (# H before it)
// ---------------------------------------------------------------------------
__global__ void bp_rows(const int* __restrict__ elems, int n,
                        const int* __restrict__ partial, int g,
                        int* __restrict__ row_of) {
    __shared__ int wsum[THREADS / 32];
    const int tid = threadIdx.x;
    const int i = blockIdx.x * THREADS + tid;
    const int flag = (i < n && elems[i] == 1) ? 1 : 0;
    const unsigned m = (unsigned)__ballot(flag != 0);
    const int wexcl = __popc(m & lane_mask_lt());
    if ((tid & 31) == 0) wsum[tid >> 5] = __popc(m);
    __syncthreads();
    if (tid == 0) {  // scan the 8 wave totals
        int acc = 0;
        for (int w = 0; w < THREADS / 32; ++w) {
            const int t = wsum[w];
            wsum[w] = acc;
            acc += t;
        }
    }
    __syncthreads();
    if (i < n) {
        const int hp = partial[blockIdx.x] + wsum[tid >> 5] + wexcl;  // excl H rank
        const int nH = partial[g];
        row_of[i] = flag ? hp : (nH + i - hp);
    }
}

// ---------------------------------------------------------------------------
// Pass 4 (specialized): scaled scatter-copy, CPR = C0+C1 chunks/row constexpr.
// Async path: stage the block's tile global->LDS with ASYNCcnt-tracked B128
// async loads, drain with s_wait_asynccnt + barrier, then scaled B128 stores.
// ---------------------------------------------------------------------------
template <int C0, int C1>
__global__ void bp_scatter_tpl(const float4* __restrict__ fp0,
                               const float4* __restrict__ fp1,
                               const int* __restrict__ row_of,
                               float4* __restrict__ out,
                               int n_atoms,
                               float s0, float b0, float s1, float b1) {
    constexpr int CPR = C0 + C1;
    const int base = blockIdx.x * TILE_ATOMS;
    const int chunks = TILE_ATOMS * CPR;

#if HAVE_ASYNC_LDS
    __shared__ float4 tile[TILE_ATOMS * CPR];
    // Stage: per-lane async B128 copies, no VGPR landing.
    for (int c = threadIdx.x; c < chunks; c += blockDim.x) {
        const int a = c / CPR;
        const int q = c - a * CPR;
        const int atom = base + a;
        if (atom < n_atoms) {
            const float4* src = (q < C0) ? (fp0 + atom * C0 + q)
                                         : (fp1 + atom * C1 + (q - C0));
            // Generic shared pointer -> 32-bit LDS address (flat->LDS mapping
            // discards upper bits, ISA 10.2), then to AS3 int4*.
            __builtin_amdgcn_global_load_async_to_lds_b128(
                (gv4i_p)(unsigned long long)src,
                (lv4i_p)(unsigned)(unsigned long long)&tile[c],
                0, 0);
        }
    }
    __builtin_amdgcn_s_wait_asynccnt(0);
    __syncthreads();
    // Drain: scale in registers, scattered B128 stores.
    for (int c = threadIdx.x; c < chunks; c += blockDim.x) {
        const int a = c / CPR;
        const int q = c - a * CPR;
        const int atom = base + a;
        if (atom < n_atoms) {
            const float4 v = tile[c];
            const float s = (q < C0) ? s0 : s1;
            const float b = (q < C0) ? b0 : b1;
            float4 r;
            r.x = fmaf(v.x, s, b);
            r.y = fmaf(v.y, s, b);
            r.z = fmaf(v.z, s, b);
            r.w = fmaf(v.w, s, b);
            out[(long)row_of[atom] * CPR + q] = r;
        }
    }
#else
    // Direct path: B128 load -> fma -> B128 store, with gfx1250 prefetch.
    for (int c = threadIdx.x; c < chunks; c += blockDim.x) {
        const int a = c / CPR;
        const int q = c - a * CPR;
        const int atom = base + a;
        if (atom >= n_atoms) continue;
        if (q == 0) {
            const int pa = atom + 2 * TILE_ATOMS;
            if (pa < n_atoms) {
                __builtin_prefetch((const void*)(fp0 + pa * C0), 0, 0);
                __builtin_prefetch((const void*)(fp1 + pa * C1), 0, 0);
            }
        }
        float4 v;
        float s, b;
        if (q < C0) {
            v = fp0[atom * C0 + q];
            s = s0; b = b0;
        } else {
            v = fp1[atom * C1 + (q - C0)];
            s = s1; b = b1;
        }
        float4 r;
        r.x = fmaf(v.x, s, b);
        r.y = fmaf(v.y, s, b);
        r.z = fmaf(v.z, s, b);
        r.w = fmaf(v.w, s, b);
        out[(long)row_of[atom] * CPR + q] = r;
    }
#endif
}

// ---------------------------------------------------------------------------
// Pass 4 (generic runtime fallback for unexpected nb0/nb1).
// ---------------------------------------------------------------------------
__global__ void bp_scatter_generic(const float4* __restrict__ fp0,
                                   const float4* __restrict__ fp1,
                                   const int* __restrict__ row_of,
                                   float4* __restrict__ out,
                                   int n_atoms, int c0, int c1,
                                   float s0, float b0, float s1, float b1) {
    const int cpr = c0 + c1;
    const int total = n_atoms * cpr;
    const int stride = gridDim.x * blockDim.x;
    for (int c = blockIdx.x * blockDim.x + threadIdx.x; c < total; c += stride) {
        const int atom = c / cpr;
        const int q = c - atom * cpr;
        float4 v;
        float s, b;
        if (q < c0) {
            v = fp0[atom * c0 + q];
            s = s0; b = b0;
        } else {
            v = fp1[atom * c1 + (q - c0)];
            s = s1; b = b1;
        }
        float4 r;
        r.x = fmaf(v.x, s, b);
        r.y = fmaf(v.y, s, b);
        r.z = fmaf(v.z, s, b);
        r.w = fmaf(v.w, s, b);
        out[(long)row_of[atom] * cpr + q] = r;
    }
}

// ---------------------------------------------------------------------------
extern "C" void kernel_launch(void* const* d_in, const int* in_sizes, int n_in,
                              void* d_out, int out_size, void* d_ws, size_t ws_size,
                              hipStream_t stream) {
    // setup_inputs() order: diff, fp_0, fp_1, jacob_0, jacob_1, elems,
    //                       jacob_ind_0, jacob_ind_1
    const float* fp0 = (const float*)d_in[1];
    const float* fp1 = (const float*)d_in[2];
    const int* elems = (const int*)d_in[5];
    float* out = (float*)d_out;

    const int n_atoms = in_sizes[5];
    const int nb0 = in_sizes[1] / n_atoms;  // 16
    const int nb1 = in_sizes[2] / n_atoms;  // 32
    const int c0 = nb0 / 4;                 // float4 chunks from fp_0
    const int c1 = nb1 / 4;                 // float4 chunks from fp_1

    // Workspace: row_of[n_atoms] ints, then partial[g+1] ints (~201 KB).
    int* row_of = (int*)d_ws;
    const int g = (n_atoms + THREADS - 1) / THREADS;
    int* partial = row_of + n_atoms;

    bp_count<<<g, THREADS, 0, stream>>>(elems, n_atoms, partial);
    bp_scan_partials<<<1, THREADS, 0, stream>>>(partial, g);
    bp_rows<<<g, THREADS, 0, stream>>>(elems, n_atoms, partial, g, row_of);

    // Static FP_RANGE from the reference:
    //  block0: lo=0,  hi=2 -> scale=1.0, bias=-1.0
    //  block1: lo=-1, hi=3 -> scale=0.5, bias=-0.5
    if (c0 == 4 && c1 == 8) {
        const int blocks = (n_atoms + TILE_ATOMS - 1) / TILE_ATOMS;
        bp_scatter_tpl<4, 8><<<blocks, THREADS, 0, stream>>>(
            (const float4*)fp0, (const float4*)fp1, row_of, (float4*)out,
            n_atoms, 1.0f, -1.0f, 0.5f, -0.5f);
    } else {
        const int total = n_atoms * (c0 + c1);
        const int blocks = (total + THREADS - 1) / THREADS;
        bp_scatter_generic<<<blocks, THREADS, 0, stream>>>(
            (const float4*)fp0, (const float4*)fp1, row_of, (float4*)out,
            n_atoms, c0, c1, 1.0f, -1.0f, 0.5f, -0.5f);
    }
}